// HierarchicalGaussianLoss_67834713473713
// MI455X (gfx1250) — compile-verified
//
#include <hip/hip_runtime.h>
#include <stdint.h>

// ---------------------------------------------------------------------------
// HierarchicalGaussianLoss for MI455X (gfx1250, wave32)
//   B=16, K=512.  Pairwise 3x3-SPD Gaussian overlap, closed form:
//   overlap = (2pi)^-1.5 * det(S)^-1/2 * exp(-0.5 * d^T adj(S) d / det(S))
// ---------------------------------------------------------------------------

#define B_DIM 16
#define K_DIM 512
#define TI 256                                   // i's per block (8 wave32)
#define TJ 128                                   // j tile staged in LDS
#define NTILE_I (K_DIM / TI)                     // 2
#define NTILE_J (K_DIM / TJ)                     // 4
#define NBLK_PAIR (B_DIM * NTILE_I * NTILE_J)    // 128

#define JSTRIDE 12                               // floats per j record (48B, 16B-aligned)

#define EPS_F 1e-6f
#define GAUSS_NORM_3D 0.063493635934240969f      // (2*pi)^{-1.5}

// --- CDNA5 async global->LDS staging (ASYNCcnt path), with safe fallback ---
#if defined(__has_builtin)
#if __has_builtin(__builtin_amdgcn_global_load_async_to_lds_b32)
#define USE_ASYNC_LDS 1
#endif
#endif

typedef __attribute__((address_space(1))) int global_i32;
typedef __attribute__((address_space(3))) int lds_i32;

__device__ __forceinline__ void stage_f32(const float* g, float* l) {
#ifdef USE_ASYNC_LDS
  // addrspacecast generic->AS1 / generic->AS3 (correct LDS offset conversion).
  __builtin_amdgcn_global_load_async_to_lds_b32(
      (global_i32*)g, (lds_i32*)l, /*offset=*/0, /*cpol=*/0);
#else
  *l = *g;
#endif
}

__device__ __forceinline__ void stage_wait() {
#ifdef USE_ASYNC_LDS
#if __has_builtin(__builtin_amdgcn_s_wait_asynccnt)
  __builtin_amdgcn_s_wait_asynccnt(0);
#else
  asm volatile("s_wait_asynccnt 0" ::: "memory");
#endif
#endif
}

// ---------------------------------------------------------------------------
// Pairwise overlap partial sums. One block = (batch, i-tile, j-tile).
// LDS record per j (12 floats, 48B, 16B-aligned):
//   [mu0 mu1 mu2 mask | s00 s01 s02 s11 | s12 s22 pad pad]
// Inner-loop reads are lane-uniform broadcasts: ds_load_b128 x2 + ds_load_b64.
// ---------------------------------------------------------------------------
__global__ __launch_bounds__(TI) void hg_pair_kernel(
    const float* __restrict__ mu, const float* __restrict__ Sigma,
    const float* __restrict__ mask, float* __restrict__ ws) {
  __shared__ __align__(16) float sj[TJ * JSTRIDE];
  __shared__ float red[TI];

  const int blk = blockIdx.x;
  const int jt  = blk % NTILE_J;
  const int it  = (blk / NTILE_J) % NTILE_I;
  const int b   = blk / (NTILE_J * NTILE_I);

  const int i   = it * TI + threadIdx.x;   // i index within [0, K)
  const int gi  = b * K_DIM + i;
  const int j0  = jt * TJ;
  const int gj0 = b * K_DIM + j0;

  // WGP-scope prefetch (locality=3): pulls the line through all cache levels,
  // unlike locality 0 which lowered to SCOPE_SYS (GL2-only, no WGP prefetch).
  __builtin_prefetch(&Sigma[gi * 9], 0, 3);

  // ---- Stage the j tile into LDS (async, tracked by ASYNCcnt) ----
  // b32 granularity is forced by global layout: mu rows are 12B-strided and
  // Sigma rows 36B-strided, so most j's are not 8/16B-aligned in global mem.
  {
    const int t = threadIdx.x;
    if (t < TJ) {                       // threads 0..127: mu + mask for j = t
      const int gj = gj0 + t;
      float* dst = &sj[t * JSTRIDE];
      const float* m = &mu[gj * 3];
      stage_f32(m + 0, dst + 0);
      stage_f32(m + 1, dst + 1);
      stage_f32(m + 2, dst + 2);
      stage_f32(&mask[gj], dst + 3);
    } else {                            // threads 128..255: 6 sym Sigma entries
      const int jj = t - TJ;
      const int gj = gj0 + jj;
      const float* S = &Sigma[gj * 9];
      float* dst = &sj[jj * JSTRIDE + 4];
      stage_f32(S + 0, dst + 0);        // s00
      stage_f32(S + 1, dst + 1);        // s01
      stage_f32(S + 2, dst + 2);        // s02
      stage_f32(S + 4, dst + 3);        // s11
      stage_f32(S + 5, dst + 4);        // s12
      stage_f32(S + 8, dst + 5);        // s22
    }
  }

  // ---- Per-thread i data (loads overlap with async staging) ----
  const float* Si = &Sigma[gi * 9];
  const float mu0 = mu[gi * 3 + 0];
  const float mu1 = mu[gi * 3 + 1];
  const float mu2 = mu[gi * 3 + 2];
  // Pre-add the eps*I regularizer on the i side.
  const float s00 = Si[0] + EPS_F, s01 = Si[1], s02 = Si[2];
  const float s11 = Si[4] + EPS_F, s12 = Si[5], s22 = Si[8] + EPS_F;
  const float mi  = mask[gi];

  stage_wait();        // per-wave: its async LDS writes are done
  __syncthreads();     // cross-wave visibility of the staged tile

  float acc = 0.0f;
#pragma unroll 4
  for (int jj = 0; jj < TJ; ++jj) {
    const float4 r0 = *reinterpret_cast<const float4*>(&sj[jj * JSTRIDE + 0]);
    const float4 r1 = *reinterpret_cast<const float4*>(&sj[jj * JSTRIDE + 4]);
    const float2 r2 = *reinterpret_cast<const float2*>(&sj[jj * JSTRIDE + 8]);

    const float d0 = mu0 - r0.x;
    const float d1 = mu1 - r0.y;
    const float d2 = mu2 - r0.z;
    const float mj = r0.w;
    const float a00 = s00 + r1.x;
    const float a01 = s01 + r1.y;
    const float a02 = s02 + r1.z;
    const float a11 = s11 + r1.w;
    const float a12 = s12 + r2.x;
    const float a22 = s22 + r2.y;

    // Symmetric 3x3: adjugate + determinant (no LU needed).
    const float c00 = a11 * a22 - a12 * a12;
    const float c01 = a02 * a12 - a01 * a22;
    const float c02 = a01 * a12 - a02 * a11;
    const float det = a00 * c00 + a01 * c01 + a02 * c02;  // > 0 (SPD + eps)
    const float c11 = a00 * a22 - a02 * a02;
    const float c12 = a01 * a02 - a00 * a12;
    const float c22 = a00 * a11 - a01 * a01;

    const float quad = d0 * d0 * c00 + d1 * d1 * c11 + d2 * d2 * c22 +
                       2.0f * (d0 * d1 * c01 + d0 * d2 * c02 + d1 * d2 * c12);
    const float maha = quad / det;
    // v_exp_f32 transcendental pipe; ~2 ulp, negligible vs fp32 sum noise.
    const float ov = GAUSS_NORM_3D * __expf(-0.5f * maha) * rsqrtf(det);

    const float w = ((j0 + jj) == i) ? 0.0f : mj;  // exclude diagonal
    acc += w * ov;
  }
  acc *= mi;

  // Deterministic fixed-order block reduction (no float atomics).
  red[threadIdx.x] = acc;
  __syncthreads();
  for (int s = TI / 2; s > 0; s >>= 1) {
    if (threadIdx.x < (unsigned)s) red[threadIdx.x] += red[threadIdx.x + s];
    __syncthreads();
  }
  if (threadIdx.x == 0) ws[2 + blk] = red[0];
}

// ---------------------------------------------------------------------------
// sum(mask) and sum(trace(Sigma)*mask)  ->  ws[0], ws[1]
// ---------------------------------------------------------------------------
__global__ __launch_bounds__(256) void hg_compact_kernel(
    const float* __restrict__ Sigma, const float* __restrict__ mask,
    float* __restrict__ ws) {
  __shared__ float sm[256];
  __shared__ float st[256];
  float am = 0.0f, at = 0.0f;
  for (int idx = threadIdx.x; idx < B_DIM * K_DIM; idx += 256) {
    const float m = mask[idx];
    const float* S = &Sigma[idx * 9];
    am += m;
    at += (S[0] + S[4] + S[8]) * m;
  }
  sm[threadIdx.x] = am;
  st[threadIdx.x] = at;
  __syncthreads();
  for (int s = 128; s > 0; s >>= 1) {
    if (threadIdx.x < (unsigned)s) {
      sm[threadIdx.x] += sm[threadIdx.x + s];
      st[threadIdx.x] += st[threadIdx.x + s];
    }
    __syncthreads();
  }
  if (threadIdx.x == 0) {
    ws[0] = sm[0];
    ws[1] = st[0];
  }
}

// ---------------------------------------------------------------------------
// Finalize the three scalars (fixed-order, deterministic).
// ---------------------------------------------------------------------------
__global__ void hg_finalize_kernel(const float* __restrict__ ws,
                                   float* __restrict__ out) {
  if (threadIdx.x == 0 && blockIdx.x == 0) {
    const float denom = fmaxf(ws[0], 1.0f);
    const float compact = ws[1] / denom;
    float sep = 0.0f;
    for (int k = 0; k < NBLK_PAIR; ++k) sep += ws[2 + k];
    sep /= (denom * (denom + 1e-6f));
    out[0] = compact;
    out[1] = sep;
    out[2] = 0.1f * compact + 1.0f * sep;
  }
}

// ---------------------------------------------------------------------------
extern "C" void kernel_launch(void* const* d_in, const int* in_sizes, int n_in,
                              void* d_out, int out_size, void* d_ws,
                              size_t ws_size, hipStream_t stream) {
  const float* mu    = (const float*)d_in[0];   // [16,512,3]
  const float* Sigma = (const float*)d_in[1];   // [16,512,3,3]
  const float* mask  = (const float*)d_in[2];   // [16,512]
  float* out = (float*)d_out;                   // 3 floats
  float* ws  = (float*)d_ws;                    // >= (2 + NBLK_PAIR) floats

  hipLaunchKernelGGL(hg_compact_kernel, dim3(1), dim3(256), 0, stream,
                     Sigma, mask, ws);
  hipLaunchKernelGGL(hg_pair_kernel, dim3(NBLK_PAIR), dim3(TI), 0, stream,
                     mu, Sigma, mask, ws);
  hipLaunchKernelGGL(hg_finalize_kernel, dim3(1), dim3(32), 0, stream,
                     ws, out);
}